// EvULoss_78228534330115
// MI455X (gfx1250) — compile-verified
//
#include <hip/hip_runtime.h>
#include <math.h>

// EvU loss for MI455X (gfx1250, wave32).
// Memory-bound: 262MB single streaming pass (~11us floor at 23.3 TB/s HBM).
// Row sums of exp() run on the matrix pipe via V_WMMA_F32_16X16X4_F32 with an
// all-ones B matrix; since B==1, D[m][n] = sum_k A[m][k] regardless of which
// column occupies which K slot, so each lane feeds 4 CONTIGUOUS floats
// (one global_load_b128) split across two independent WMMA accumulators.
// v_exp_f32 (TRANS) co-executes with the matrix pipe.

typedef __attribute__((ext_vector_type(2))) float v2f;
typedef __attribute__((ext_vector_type(4))) float v4f;
typedef __attribute__((ext_vector_type(8))) float v8f;

#define EVU_EPS 1e-10f

// ---------------------------------------------------------------- kernel 0
__global__ void evu_init_kernel(unsigned* __restrict__ minmax) {
  minmax[0] = 0x7F800000u;  // +inf (unc > 0 -> uint order == float order)
  minmax[1] = 0x00000000u;  // 0
}

// ---------------------------------------------------------------- kernel 1
__global__ __launch_bounds__(256) void evu_rowstats_kernel(
    const float* __restrict__ x, const long long* __restrict__ target,
    float* __restrict__ uncA, float* __restrict__ maxAlphaA, int* __restrict__ corrA,
    unsigned* __restrict__ minmax, int N, int C) {
  const int lane = threadIdx.x & 31;
  const int waveInBlock = threadIdx.x >> 5;
  const int waveGlobal = blockIdx.x * (blockDim.x >> 5) + waveInBlock;
  const int rowBase = waveGlobal * 16;
  if (rowBase >= N) return;  // wave-uniform

  const int m = lane & 15;   // tile row
  const int half = lane >> 4;
  const int row = rowBase + m;

  float mmax = -__builtin_inff();
  int midx = 0;
  float sumExp = 0.0f;

  // Wave-uniform fast-path condition: all 16 rows valid, rows 16B-aligned.
  const bool fastPath = (rowBase + 16 <= N) && ((C & 3) == 0);

  if (fastPath) {
    const float* rowPtr = x + (size_t)row * (size_t)C;

    v8f acc0 = {0.f, 0.f, 0.f, 0.f, 0.f, 0.f, 0.f, 0.f};
    v8f acc1 = {0.f, 0.f, 0.f, 0.f, 0.f, 0.f, 0.f, 0.f};
    v2f bOnes;
    bOnes.x = 1.0f;
    bOnes.y = 1.0f;

    const int C8 = C & ~7;
    for (int kb = 0; kb < C8; kb += 8) {
      const int c0 = kb + 4 * half;               // lanes 0-15: kb.., 16-31: kb+4..
      v4f f = *(const v4f*)(rowPtr + c0);         // global_load_b128, 16B aligned
      if (f.x > mmax) { mmax = f.x; midx = c0; }
      if (f.y > mmax) { mmax = f.y; midx = c0 + 1; }
      if (f.z > mmax) { mmax = f.z; midx = c0 + 2; }
      if (f.w > mmax) { mmax = f.w; midx = c0 + 3; }
      v2f a0, a1;
      a0.x = __expf(f.x);
      a0.y = __expf(f.y);
      a1.x = __expf(f.z);
      a1.y = __expf(f.w);
      // Two independent accumulators -> the two WMMAs per iteration overlap.
      acc0 = __builtin_amdgcn_wmma_f32_16x16x4_f32(false, a0, false, bOnes,
                                                   (short)0, acc0, false, false);
      acc1 = __builtin_amdgcn_wmma_f32_16x16x4_f32(false, a1, false, bOnes,
                                                   (short)0, acc1, false, false);
    }

    // Combine max/argmax across column halves (ascending order per lane, so
    // equal-keys-pick-smaller-index preserves argmax-first semantics).
    {
      float omax = __shfl_xor(mmax, 16, 32);
      int oidx = __shfl_xor(midx, 16, 32);
      if (omax > mmax || (omax == mmax && oidx < midx)) { mmax = omax; midx = oidx; }
    }

    // Extract row sum from D: VGPR r holds M=r (lanes 0-15) / M=r+8 (lanes 16-31);
    // all N columns are identical because B == ones.
    float cf[8];
#pragma unroll
    for (int i = 0; i < 8; ++i) cf[i] = acc0[i] + acc1[i];
    float sel = cf[lane & 7];  // lane s holds D[(s&7) + 8*(s>=16)][s&15]
    const int srcLane = (m & 7) | ((m >> 3) << 4);
    sumExp = __shfl(sel, srcLane, 32);

    if (half == 0) {
      // Tail columns (C%8), scalar.
      for (int c = C8; c < C; ++c) {
        float xv = rowPtr[c];
        sumExp += __expf(xv);
        if (xv > mmax) { mmax = xv; midx = c; }
      }
      const float fC = (float)C;
      const float unc = fC / (sumExp + fC);        // C / sum(exp(x)+1)
      uncA[row] = unc;
      maxAlphaA[row] = __expf(mmax) + 1.0f;        // max(alpha)
      corrA[row] = (midx == (int)target[row]) ? 1 : 0;
      atomicMin(&minmax[0], __float_as_uint(unc));
      atomicMax(&minmax[1], __float_as_uint(unc));
    }
  } else {
    // Generic fallback (ragged N tile or unaligned C): lanes 0-15 scan scalar.
    if (half == 0 && row < N) {
      const float* rowPtr = x + (size_t)row * (size_t)C;
      for (int c = 0; c < C; ++c) {
        float xv = rowPtr[c];
        sumExp += __expf(xv);
        if (xv > mmax) { mmax = xv; midx = c; }
      }
      const float fC = (float)C;
      const float unc = fC / (sumExp + fC);
      uncA[row] = unc;
      maxAlphaA[row] = __expf(mmax) + 1.0f;
      corrA[row] = (midx == (int)target[row]) ? 1 : 0;
      atomicMin(&minmax[0], __float_as_uint(unc));
      atomicMax(&minmax[1], __float_as_uint(unc));
    }
  }
}

// ---------------------------------------------------------------- kernel 2
__global__ __launch_bounds__(256) void evu_classify_kernel(
    const float* __restrict__ uncA, const float* __restrict__ maxAlphaA,
    const int* __restrict__ corrA, const unsigned* __restrict__ minmax,
    const float* __restrict__ optTh, float* __restrict__ partials, int N) {
  __shared__ float sdata[4][256];
  const float umin = __uint_as_float(minmax[0]);
  const float umax = __uint_as_float(minmax[1]);
  const float th = umin + optTh[0] * (umax - umin);

  const int i = blockIdx.x * 256 + (int)threadIdx.x;
  float v0 = 0.f, v1 = 0.f, v2 = 0.f, v3 = 0.f;
  if (i < N) {
    const float unc = uncA[i];
    const float ma = maxAlphaA[i];
    const int corr = corrA[i];
    const float t = tanhf(unc);
    const bool certain = (unc <= th);
    if (corr) {
      if (certain) v0 = ma * (1.0f - t);
      else         v1 = ma * t;
    } else {
      if (certain) v2 = (1.0f - ma) * (1.0f - t);
      else         v3 = (1.0f - ma) * t;
    }
  }
  sdata[0][threadIdx.x] = v0;
  sdata[1][threadIdx.x] = v1;
  sdata[2][threadIdx.x] = v2;
  sdata[3][threadIdx.x] = v3;
  __syncthreads();
  for (int s = 128; s > 0; s >>= 1) {
    if ((int)threadIdx.x < s) {
#pragma unroll
      for (int j = 0; j < 4; ++j)
        sdata[j][threadIdx.x] += sdata[j][threadIdx.x + s];
    }
    __syncthreads();
  }
  if (threadIdx.x == 0) {
#pragma unroll
    for (int j = 0; j < 4; ++j) partials[blockIdx.x * 4 + j] = sdata[j][0];
  }
}

// ---------------------------------------------------------------- kernel 3
__global__ __launch_bounds__(256) void evu_finalize_kernel(
    const float* __restrict__ partials, int numPartials, float* __restrict__ out) {
  __shared__ float sdata[4][256];
  float v[4] = {0.f, 0.f, 0.f, 0.f};
  for (int i = threadIdx.x; i < numPartials; i += 256) {
#pragma unroll
    for (int j = 0; j < 4; ++j) v[j] += partials[i * 4 + j];
  }
#pragma unroll
  for (int j = 0; j < 4; ++j) sdata[j][threadIdx.x] = v[j];
  __syncthreads();
  for (int s = 128; s > 0; s >>= 1) {
    if ((int)threadIdx.x < s) {
#pragma unroll
      for (int j = 0; j < 4; ++j)
        sdata[j][threadIdx.x] += sdata[j][threadIdx.x + s];
    }
    __syncthreads();
  }
  if (threadIdx.x == 0) {
    const float n_ac = sdata[0][0];
    const float n_au = sdata[1][0];
    const float n_ic = sdata[2][0];
    const float n_iu = sdata[3][0];
    const float evu = (n_ac + n_iu) / (n_ac + n_au + n_ic + n_iu + EVU_EPS);
    out[0] = -logf(evu + EVU_EPS);  // BETA = 1.0
  }
}

// ---------------------------------------------------------------- launch
extern "C" void kernel_launch(void* const* d_in, const int* in_sizes, int n_in,
                              void* d_out, int out_size, void* d_ws, size_t ws_size,
                              hipStream_t stream) {
  const float* x = (const float*)d_in[0];               // [N, C] float32
  const long long* target = (const long long*)d_in[1];  // [N] int64
  const float* optTh = (const float*)d_in[2];           // [1] float32
  // d_in[3] (num_classes) == C, recovered from shapes.

  const int N = in_sizes[1];
  const int C = in_sizes[0] / N;

  char* ws = (char*)d_ws;
  float* uncA = (float*)ws;                              // N floats
  float* maxAlphaA = (float*)(ws + (size_t)N * 4);       // N floats
  int* corrA = (int*)(ws + (size_t)N * 8);               // N ints
  unsigned* minmax = (unsigned*)(ws + (size_t)N * 12);   // 2 uints (+pad)
  float* partials = (float*)(ws + (size_t)N * 12 + 16);  // blocks2*4 floats

  evu_init_kernel<<<1, 1, 0, stream>>>(minmax);

  const int rowsPerBlock = 128;  // 8 waves * 16 rows
  const int blocks1 = (N + rowsPerBlock - 1) / rowsPerBlock;
  evu_rowstats_kernel<<<blocks1, 256, 0, stream>>>(x, target, uncA, maxAlphaA,
                                                   corrA, minmax, N, C);

  const int blocks2 = (N + 255) / 256;
  evu_classify_kernel<<<blocks2, 256, 0, stream>>>(uncA, maxAlphaA, corrA, minmax,
                                                   optTh, partials, N);

  evu_finalize_kernel<<<1, 256, 0, stream>>>(partials, blocks2, (float*)d_out);
}